// GNNModel_18245021073918
// MI455X (gfx1250) — compile-verified
//
#include <hip/hip_runtime.h>
#include <hip/hip_bf16.h>

typedef float v2f __attribute__((ext_vector_type(2)));
typedef float v8f __attribute__((ext_vector_type(8)));

#define EMB 128

// ---------------- utility ----------------
__global__ void zero_kernel(float* __restrict__ p, unsigned n) {
    unsigned i = blockIdx.x * blockDim.x + threadIdx.x;
    if (i < n) p[i] = 0.0f;
}

// deg[dst] += 1 per edge (float atomics; counts are exact up to 2^24)
__global__ void deg_kernel(const long long* __restrict__ dst, float* __restrict__ deg, int E) {
    int e = blockIdx.x * blockDim.x + threadIdx.x;
    if (e < E) atomicAdd(&deg[dst[e]], 1.0f);
}

// in-place: deg -> rsqrt(deg + 1)   (self-loop added)
__global__ void dis_kernel(float* __restrict__ degdis, int n) {
    int i = blockIdx.x * blockDim.x + threadIdx.x;
    if (i < n) degdis[i] = rsqrtf(degdis[i] + 1.0f);
}

// ---------------- fp32 WMMA GEMM:  C[N x NOUT] = A[N x K] @ W[K x NOUT] ----------------
// One wave per 16x16 output tile; blockDim.x = 32 * (NOUT/16); grid = N/16 (N % 16 == 0).
// W is staged in LDS (W1: 32KB, W2: 8KB of the 320KB WGP LDS).
template <int K, int NOUT>
__global__ void gemm_wmma_f32(const float* __restrict__ A,
                              const float* __restrict__ W,
                              float* __restrict__ C) {
    __shared__ float sW[K * NOUT];
    const int tid = threadIdx.x;
    for (int i = tid; i < K * NOUT; i += blockDim.x) sW[i] = W[i];
    __syncthreads();

    const int wave = tid >> 5;
    const int lane = tid & 31;
    const int half = lane >> 4;   // 0: k+0/k+1, 1: k+2/k+3 (A) / same for B
    const int idx  = lane & 15;   // m for A-frag, n for B-frag, n for C/D
    const int m0   = blockIdx.x * 16;
    const int n0   = wave * 16;
    const int kb   = half * 2;

    const float2* arow = reinterpret_cast<const float2*>(A + (size_t)(m0 + idx) * K);

    v8f acc = {0.f, 0.f, 0.f, 0.f, 0.f, 0.f, 0.f, 0.f};
#pragma unroll
    for (int k = 0; k < K; k += 4) {
        float2 av = arow[(k + kb) >> 1];          // A[m][k+kb], A[m][k+kb+1]
        v2f a; a[0] = av.x; a[1] = av.y;
        v2f b;
        b[0] = sW[(k + kb) * NOUT + n0 + idx];    // B[k+kb][n]
        b[1] = sW[(k + kb + 1) * NOUT + n0 + idx];
        // 8 args: (neg_a, A, neg_b, B, c_mod, C, reuse_a, reuse_b)
        acc = __builtin_amdgcn_wmma_f32_16x16x4_f32(false, a, false, b,
                                                    (short)0, acc, false, false);
    }

#pragma unroll
    for (int v = 0; v < 8; ++v) {
        int m = v + half * 8;
        C[(size_t)(m0 + m) * NOUT + n0 + idx] = acc[v];
    }
}

// ---------------- edge scatter:  agg[dst] += xw[src] * (dis[src]*dis[dst]) ----------------
// F/2 threads per edge, 2 features (float2) per thread.
template <int F>
__global__ void scatter_kernel(const float* __restrict__ xw,
                               const float* __restrict__ dis,
                               const long long* __restrict__ src,
                               const long long* __restrict__ dst,
                               float* __restrict__ agg, int E) {
    constexpr int TPE = F / 2;                     // power of 2
    unsigned gid = blockIdx.x * blockDim.x + threadIdx.x;
    unsigned e = gid / TPE;
    unsigned j = gid % TPE;
    if (e >= (unsigned)E) return;
    long long s = src[e];
    long long d = dst[e];
    float norm = dis[s] * dis[d];
    float2 v = reinterpret_cast<const float2*>(xw + s * F)[j];
    float* out = agg + d * F + j * 2;
    atomicAdd(out,     v.x * norm);
    atomicAdd(out + 1, v.y * norm);
}

// ---------------- self-loop + bias (+ optional ReLU), in place on agg ----------------
template <int F, bool RELU>
__global__ void selfloop_kernel(float* __restrict__ agg,
                                const float* __restrict__ xw,
                                const float* __restrict__ dis,
                                const float* __restrict__ b, int n) {
    unsigned gid = blockIdx.x * blockDim.x + threadIdx.x;
    if (gid >= (unsigned)n * F) return;
    unsigned i = gid / F;
    unsigned f = gid % F;
    float di = dis[i];
    float v = agg[gid] + di * di * xw[gid] + b[f];
    agg[gid] = RELU ? fmaxf(v, 0.0f) : v;
}

// ---------------- decoder: one wave32 per edge, lane == feature (32 dims) ----------------
__global__ void decode_kernel(const float* __restrict__ z,
                              const long long* __restrict__ src,
                              const long long* __restrict__ dst,
                              float* __restrict__ out, int E) {
    int wave = threadIdx.x >> 5;
    int lane = threadIdx.x & 31;
    long long e = (long long)blockIdx.x * (blockDim.x >> 5) + wave;
    if (e >= E) return;
    long long s = src[e];
    long long d = dst[e];
    float val = z[s * 32 + lane] * z[d * 32 + lane];
#pragma unroll
    for (int off = 16; off > 0; off >>= 1)
        val += __shfl_xor(val, off, 32);
    if (lane == 0) out[e] = 1.0f / (1.0f + expf(-val));
}

// ---------------- launcher ----------------
extern "C" void kernel_launch(void* const* d_in, const int* in_sizes, int n_in,
                              void* d_out, int out_size, void* d_ws, size_t ws_size,
                              hipStream_t stream) {
    const float*     emb = (const float*)d_in[0];   // [N,128]
    const float*     W1  = (const float*)d_in[1];   // [128,64]
    const float*     b1  = (const float*)d_in[2];   // [64]
    const float*     W2  = (const float*)d_in[3];   // [64,32]
    const float*     b2  = (const float*)d_in[4];   // [32]
    const long long* ei  = (const long long*)d_in[5]; // [2,E] int64

    const int N = in_sizes[0] / EMB;
    const int E = in_sizes[5] / 2;
    const long long* src = ei;
    const long long* dst = ei + E;

    // workspace layout (floats), 256B-aligned chunks
    float* ws = (float*)d_ws;
    const size_t Na = (size_t)((N + 63) & ~63);
    float* degdis = ws;                       // N       (deg -> rsqrt(deg+1))
    float* xw1    = degdis + Na;              // N*64
    float* agg1   = xw1 + (size_t)N * 64;     // N*64    (becomes h)
    float* xw2    = agg1 + (size_t)N * 64;    // N*32
    float* agg2   = xw2 + (size_t)N * 32;     // N*32    (becomes z)
    float* outp   = (float*)d_out;            // E

    const int TB = 256;
    auto blocks = [&](long long total) { return (unsigned)((total + TB - 1) / TB); };

    // zero accumulators (must re-zero every call: determinism)
    zero_kernel<<<blocks(N), TB, 0, stream>>>(degdis, (unsigned)N);
    zero_kernel<<<blocks((long long)N * 64), TB, 0, stream>>>(agg1, (unsigned)(N * 64));
    zero_kernel<<<blocks((long long)N * 32), TB, 0, stream>>>(agg2, (unsigned)(N * 32));

    // degree + symmetric normalization
    deg_kernel<<<blocks(E), TB, 0, stream>>>(dst, degdis, E);
    dis_kernel<<<blocks(N), TB, 0, stream>>>(degdis, N);

    // layer 1: xw1 = X @ W1  (fp32 WMMA), scatter, self-loop + bias + ReLU
    gemm_wmma_f32<128, 64><<<N / 16, 128, 0, stream>>>(emb, W1, xw1);
    scatter_kernel<64><<<blocks((long long)E * 32), TB, 0, stream>>>(xw1, degdis, src, dst, agg1, E);
    selfloop_kernel<64, true><<<blocks((long long)N * 64), TB, 0, stream>>>(agg1, xw1, degdis, b1, N);

    // layer 2: xw2 = h @ W2  (fp32 WMMA), scatter, self-loop + bias
    gemm_wmma_f32<64, 32><<<N / 16, 64, 0, stream>>>(agg1, W2, xw2);
    scatter_kernel<32><<<blocks((long long)E * 16), TB, 0, stream>>>(xw2, degdis, src, dst, agg2, E);
    selfloop_kernel<32, false><<<blocks((long long)N * 32), TB, 0, stream>>>(agg2, xw2, degdis, b2, N);

    // decoder: sigmoid(dot(z[src], z[dst])) — one wave per edge
    decode_kernel<<<(E + 7) / 8, TB, 0, stream>>>(agg2, src, dst, outp, E);
}